// MSA_Conv2d_v1_89773406421554
// MI455X (gfx1250) — compile-verified
//
#include <hip/hip_runtime.h>

typedef __attribute__((ext_vector_type(2))) float v2f;
typedef __attribute__((ext_vector_type(8))) float v8f;

// Problem dims (fixed by the reference)
constexpr int Bn  = 8;
constexpr int Cn  = 128;
constexpr int Hn  = 128;
constexpr int Wn  = 128;
constexpr int HW  = Hn * Wn;        // 16384
constexpr int CHW = Cn * HW;        // 2097152
constexpr int NSH = 9;              // 3x3 shifts

// One wave32 per (b, h, 16-wide w tile).
// A[m][k] = Q[b, k, h, w0+m]          (M=16 pixels, K=128 channels)
// B[k][n] = Kt[b, k, h+di, w0+n+dj]   (zero outside image)
// D = A x B ; out[b, s, h, w0+m] = D[m][m]
__global__ __launch_bounds__(256)
void msa_conv2d_wmma_f32(const float* __restrict__ Q,
                         const float* __restrict__ Kt,
                         float* __restrict__ out)
{
    const int lane = threadIdx.x & 31;
    const int wave = threadIdx.x >> 5;
    const int tile = blockIdx.x * 8 + wave;     // 0 .. 8191
    const int tw   = tile & 7;                  // w tile
    const int h    = (tile >> 3) & 127;
    const int b    = tile >> 10;
    const int w0   = tw * 16;

    // ISA 16x4 f32 A layout: lanes 0-15 -> M=lane, K={0,1}; lanes 16-31 -> M=lane-16, K={2,3}
    const int m     = lane & 15;
    const int khalf = (lane >> 4) << 1;         // 0 or 2

    // ---- Preload A (Q tile) into registers: read Q exactly once ----
    const float* qbase = Q + (size_t)b * CHW + (size_t)h * Wn + (w0 + m);
    v2f A[32];
#pragma unroll
    for (int kc = 0; kc < 32; ++kc) {
        const int c0 = kc * 4 + khalf;
        A[kc].x = qbase[(size_t)c0 * HW];        // K = c0
        A[kc].y = qbase[(size_t)(c0 + 1) * HW];  // K = c0+1
    }

    const size_t obase = (size_t)b * NSH * HW + (size_t)h * Wn + w0;

#pragma unroll 1
    for (int s = 0; s < NSH; ++s) {
        const int di = s / 3 - 1;
        const int dj = s % 3 - 1;
        const int hh = h + di;
        const int ww = w0 + m + dj;             // B column N = m (mirrors A lane layout)
        const bool valid = ((unsigned)hh < (unsigned)Hn) && ((unsigned)ww < (unsigned)Wn);
        // Clamp so the load itself is always in-bounds; zero via select afterwards.
        const int hhc = hh < 0 ? 0 : (hh > Hn - 1 ? Hn - 1 : hh);
        const int wwc = ww < 0 ? 0 : (ww > Wn - 1 ? Wn - 1 : ww);
        const float* kbase = Kt + (size_t)b * CHW + (size_t)hhc * Wn + wwc;

        v8f acc = {};
#pragma unroll
        for (int kc = 0; kc < 32; ++kc) {
            const int c0 = kc * 4 + khalf;
            const float bx = kbase[(size_t)c0 * HW];
            const float by = kbase[(size_t)(c0 + 1) * HW];
            v2f Bv;
            Bv.x = valid ? bx : 0.0f;
            Bv.y = valid ? by : 0.0f;
            // (neg_a, A, neg_b, B, c_mod, C, reuse_a, reuse_b)
            acc = __builtin_amdgcn_wmma_f32_16x16x4_f32(
                false, A[kc], false, Bv, (short)0, acc, false, false);
        }

        // Diagonal extraction. C/D layout: VGPR r, lanes 0-15 -> (M=r, N=lane),
        // lanes 16-31 -> (M=r+8, N=lane-16).  D[m][m] lives in:
        //   m in 0..7  : lane = m,      vgpr r = m
        //   m in 8..15 : lane = m + 16, vgpr r = m - 8
        const int r = lane & 7;
        float v = acc[0];
        v = (r == 1) ? acc[1] : v;
        v = (r == 2) ? acc[2] : v;
        v = (r == 3) ? acc[3] : v;
        v = (r == 4) ? acc[4] : v;
        v = (r == 5) ? acc[5] : v;
        v = (r == 6) ? acc[6] : v;
        v = (r == 7) ? acc[7] : v;

        if (lane < 8 || lane >= 24) {
            const int mm = (lane < 8) ? lane : lane - 16;
            out[obase + (size_t)s * HW + mm] = v;
        }
    }
}

extern "C" void kernel_launch(void* const* d_in, const int* in_sizes, int n_in,
                              void* d_out, int out_size, void* d_ws, size_t ws_size,
                              hipStream_t stream)
{
    const float* Q  = (const float*)d_in[0];
    const float* Kt = (const float*)d_in[1];
    float* out      = (float*)d_out;

    // 8192 waves = B * H * (W/16); 8 waves per 256-thread block
    const int blocks = (Bn * Hn * (Wn / 16)) / 8;   // 1024
    msa_conv2d_wmma_f32<<<blocks, 256, 0, stream>>>(Q, Kt, out);
}